// TripletLossWithMining_65498251264446
// MI455X (gfx1250) — compile-verified
//
#include <hip/hip_runtime.h>

typedef __attribute__((ext_vector_type(2))) float v2f;
typedef __attribute__((ext_vector_type(8))) float v8f;

#define DIM 128
#define NSTEP (DIM / 4)    // 32 k-steps of V_WMMA_F32_16X16X4_F32
#define MARGIN 1.0f
#define CCHUNKS 4          // column chunks (grid.y)
#define LDS_STRIDE 132     // dword stride per column: 132 mod 64 == 4 -> conflict-free b64 reads

// ---- gfx1250 async global->LDS path (ASYNCcnt), with sync fallback --------
#if defined(__has_builtin)
#if __has_builtin(__builtin_amdgcn_global_load_async_to_lds_b64) && \
    __has_builtin(__builtin_amdgcn_s_wait_asynccnt)
#define USE_ASYNC_LDS 1
#endif
#endif
#ifndef USE_ASYNC_LDS
#define USE_ASYNC_LDS 0
#endif

#if USE_ASYNC_LDS
// builtin signature (from hipcc diagnostic): payload type is int __vector(2)
typedef int v2i_b64 __attribute__((vector_size(8)));
typedef __attribute__((address_space(1))) v2i_b64* gb64_t;
typedef __attribute__((address_space(3))) v2i_b64* lb64_t;

__device__ __forceinline__ void async_copy_b64(const float* g, float* lds) {
  __builtin_amdgcn_global_load_async_to_lds_b64((gb64_t)g, (lb64_t)lds, 0, 0);
}
#endif

// ---------------------------------------------------------------------------
// Kernel 1: row squared-norms, init hardest-neg to +inf, zero accumulators
// ---------------------------------------------------------------------------
__global__ void __launch_bounds__(256)
sq_init_kernel(const float* __restrict__ E, float* __restrict__ sq,
               unsigned* __restrict__ hardestU, float* __restrict__ total,
               unsigned* __restrict__ count, int B) {
  int i = blockIdx.x * blockDim.x + threadIdx.x;
  if (i < B) {
    const float4* r = (const float4*)(E + (size_t)i * DIM);
    float s = 0.f;
#pragma unroll
    for (int k = 0; k < DIM / 4; ++k) {
      float4 v = r[k];
      s += v.x * v.x + v.y * v.y + v.z * v.z + v.w * v.w;
    }
    sq[i] = s;
    hardestU[i] = 0x7F800000u;  // +inf bits; dist >= 0 so uint order == float order
  }
  if (i == 0) { *total = 0.f; *count = 0u; }
}

// ---------------------------------------------------------------------------
// Kernel 2: hardest-negative mining, fused fp32 WMMA Gram tiles.
// Block = 128 anchor rows (8 waves x 16 rows), all waves share one column
// stream staged in double-buffered LDS. Never materializes the dist matrix.
// ---------------------------------------------------------------------------
__global__ void __launch_bounds__(256)
hardest_neg_kernel(const float* __restrict__ E, const int* __restrict__ labels,
                   const float* __restrict__ sq, unsigned* __restrict__ hardestU,
                   int B) {
  __shared__ float bt[2][16][LDS_STRIDE];   // 2 x 16 cols x 128 dims (+pad)

  const int lane = threadIdx.x & 31;
  const int wave = threadIdx.x >> 5;
  const int lo = lane & 15;   // column within tile / A-row within tile
  const int hi = lane >> 4;   // K-pair selector per ISA fragment layout
  const int rowBase = blockIdx.x * 128 + wave * 16;

  // ---- A fragments, resident in VGPRs for the whole sweep -----------------
  // lane covers row rowBase+lo, K pair {4s+2hi, 4s+2hi+1}
  const float* arow = E + (size_t)(rowBase + lo) * DIM + 2 * hi;
  v2f a[NSTEP];
#pragma unroll
  for (int s = 0; s < NSTEP; ++s)
    a[s] = *(const v2f*)(arow + 4 * s);

  // C/D layout: lane L, vgpr v -> row = rowBase + 8*hi + v, col = jt + lo
  float sqRow[8];
  int labRow[8];
#pragma unroll
  for (int v = 0; v < 8; ++v) {
    int r = rowBase + 8 * hi + v;
    sqRow[v] = sq[r];
    labRow[v] = labels[r];
  }

  float mn[8];
#pragma unroll
  for (int v = 0; v < 8; ++v) mn[v] = __builtin_inff();

  const int chunk = B / CCHUNKS;
  const int jBegin = blockIdx.y * chunk;
  const int jEnd = jBegin + chunk;

  // Cooperative fill: thread covers float4 #lane of columns (wave) and (wave+8)
  const int fc0 = wave;        // column 0..7
  const int fc1 = wave + 8;    // column 8..15
  const int fq = lane;         // float4 index 0..31 within the 128-dim column

  // ---- prologue: fill buffer 0 with tile jBegin ---------------------------
  {
    const float* g0 = E + (size_t)(jBegin + fc0) * DIM + 4 * fq;
    const float* g1 = E + (size_t)(jBegin + fc1) * DIM + 4 * fq;
#if USE_ASYNC_LDS
    async_copy_b64(g0 + 0, &bt[0][fc0][4 * fq + 0]);
    async_copy_b64(g0 + 2, &bt[0][fc0][4 * fq + 2]);
    async_copy_b64(g1 + 0, &bt[0][fc1][4 * fq + 0]);
    async_copy_b64(g1 + 2, &bt[0][fc1][4 * fq + 2]);
    __builtin_amdgcn_s_wait_asynccnt(0);
#else
    float4 p0 = *(const float4*)g0;
    float4 p1 = *(const float4*)g1;
    *(v2f*)&bt[0][fc0][4 * fq + 0] = v2f{p0.x, p0.y};
    *(v2f*)&bt[0][fc0][4 * fq + 2] = v2f{p0.z, p0.w};
    *(v2f*)&bt[0][fc1][4 * fq + 0] = v2f{p1.x, p1.y};
    *(v2f*)&bt[0][fc1][4 * fq + 2] = v2f{p1.z, p1.w};
#endif
  }
  __syncthreads();

  int buf = 0;
  for (int jt = jBegin; jt < jEnd; jt += 16) {
    // -- kick off next tile's copy first (hidden behind WMMA) ---------------
    const bool hasNext = (jt + 16) < jEnd;
#if USE_ASYNC_LDS
    if (hasNext) {
      const float* g0 = E + (size_t)(jt + 16 + fc0) * DIM + 4 * fq;
      const float* g1 = E + (size_t)(jt + 16 + fc1) * DIM + 4 * fq;
      async_copy_b64(g0 + 0, &bt[buf ^ 1][fc0][4 * fq + 0]);
      async_copy_b64(g0 + 2, &bt[buf ^ 1][fc0][4 * fq + 2]);
      async_copy_b64(g1 + 0, &bt[buf ^ 1][fc1][4 * fq + 0]);
      async_copy_b64(g1 + 2, &bt[buf ^ 1][fc1][4 * fq + 2]);
    }
#else
    float4 p0, p1;
    if (hasNext) {
      p0 = *(const float4*)(E + (size_t)(jt + 16 + fc0) * DIM + 4 * fq);
      p1 = *(const float4*)(E + (size_t)(jt + 16 + fc1) * DIM + 4 * fq);
    }
#endif

    const int c0 = jt + lo;
    const float sqC = sq[c0];
    const int labC = labels[c0];

    // -- WMMA over K: B fragments from LDS, 4 independent accumulator chains
    const float* bp = &bt[buf][lo][2 * hi];
    v8f ac[4] = {};
#pragma unroll
    for (int g = 0; g < 2; ++g) {
      v2f bf[16];
#pragma unroll
      for (int s = 0; s < 16; ++s)
        bf[s] = *(const v2f*)(bp + 4 * (16 * g + s));
#pragma unroll
      for (int s = 0; s < 16; ++s)
        ac[s & 3] = __builtin_amdgcn_wmma_f32_16x16x4_f32(
            false, a[16 * g + s], false, bf[s], (short)0, ac[s & 3], false, false);
    }
    v8f acc = (ac[0] + ac[1]) + (ac[2] + ac[3]);

    // -- epilogue: distance + masked running min ----------------------------
#pragma unroll
    for (int v = 0; v < 8; ++v) {
      float dist = fmaxf(sqRow[v] + sqC - 2.0f * acc[v], 0.0f);
      bool isNeg = (labRow[v] != labC);   // diagonal is same-label -> excluded
      mn[v] = (isNeg && dist < mn[v]) ? dist : mn[v];
    }

    // -- make next tile visible ---------------------------------------------
#if USE_ASYNC_LDS
    __builtin_amdgcn_s_wait_asynccnt(0);
#else
    if (hasNext) {
      *(v2f*)&bt[buf ^ 1][fc0][4 * fq + 0] = v2f{p0.x, p0.y};
      *(v2f*)&bt[buf ^ 1][fc0][4 * fq + 2] = v2f{p0.z, p0.w};
      *(v2f*)&bt[buf ^ 1][fc1][4 * fq + 0] = v2f{p1.x, p1.y};
      *(v2f*)&bt[buf ^ 1][fc1][4 * fq + 2] = v2f{p1.z, p1.w};
    }
#endif
    __syncthreads();
    buf ^= 1;
  }

  // min across the 16 lanes of each half-wave (masks <= 8 stay in-half)
#pragma unroll
  for (int v = 0; v < 8; ++v) {
#pragma unroll
    for (int off = 8; off >= 1; off >>= 1)
      mn[v] = fminf(mn[v], __shfl_xor(mn[v], off, 32));
  }
  if (lo == 0) {
#pragma unroll
    for (int v = 0; v < 8; ++v)
      atomicMin(&hardestU[rowBase + 8 * hi + v], __float_as_uint(mn[v]));
  }
}

// ---------------------------------------------------------------------------
// Kernel 3: loss over same-label (positive) pairs only (~0.1% of pass-1 work)
// ---------------------------------------------------------------------------
__global__ void __launch_bounds__(256)
pos_loss_kernel(const float* __restrict__ E, const int* __restrict__ labels,
                const float* __restrict__ sq, const unsigned* __restrict__ hardestU,
                float* __restrict__ total, unsigned* __restrict__ count, int B) {
  __shared__ float ei[DIM];
  const int i = blockIdx.x;
  const int labI = labels[i];
  const float sqI = sq[i];
  const float hn = __uint_as_float(hardestU[i]);  // +inf => has_neg==false => nothing valid

  for (int k = threadIdx.x; k < DIM; k += blockDim.x)
    ei[k] = E[(size_t)i * DIM + k];
  __syncthreads();

  float tot = 0.f;
  unsigned cnt = 0u;
  for (int j = threadIdx.x; j < B; j += blockDim.x) {
    if (j == i || labels[j] != labI) continue;
    const float4* rj = (const float4*)(E + (size_t)j * DIM);
    float dot = 0.f;
#pragma unroll
    for (int k = 0; k < DIM / 4; ++k) {
      float4 v = rj[k];
      dot += ei[4 * k + 0] * v.x + ei[4 * k + 1] * v.y +
             ei[4 * k + 2] * v.z + ei[4 * k + 3] * v.w;
    }
    float dist = fmaxf(sqI + sq[j] - 2.0f * dot, 0.0f);
    if (hn < dist) {                    // valid triplet (hardest_neg < pos_dist)
      tot += fmaxf(dist - hn + MARGIN, 0.0f);
      cnt += 1u;
    }
  }

  // wave32 reduction, then one atomic per wave
#pragma unroll
  for (int off = 16; off >= 1; off >>= 1) {
    tot += __shfl_xor(tot, off, 32);
    cnt += __shfl_xor(cnt, off, 32);
  }
  if ((threadIdx.x & 31) == 0 && cnt) {
    atomicAdd(total, tot);
    atomicAdd(count, cnt);
  }
}

// ---------------------------------------------------------------------------
// Kernel 4: finalize -> d_out = [loss, count]
// ---------------------------------------------------------------------------
__global__ void finalize_kernel(const float* __restrict__ total,
                                const unsigned* __restrict__ count,
                                float* __restrict__ out) {
  unsigned c = *count;
  unsigned cd = c > 1u ? c : 1u;
  out[0] = *total / (float)cd;
  out[1] = (float)c;
}

// ---------------------------------------------------------------------------
extern "C" void kernel_launch(void* const* d_in, const int* in_sizes, int n_in,
                              void* d_out, int out_size, void* d_ws, size_t ws_size,
                              hipStream_t stream) {
  const float* E = (const float*)d_in[0];       // [B, 128] fp32
  const int* labels = (const int*)d_in[1];      // [B]
  const int B = in_sizes[1];                    // 8192

  // workspace layout
  float* sq = (float*)d_ws;                     // B floats
  unsigned* hardestU = (unsigned*)(sq + B);     // B uints (float bits)
  float* total = (float*)(hardestU + B);        // 1 float
  unsigned* count = (unsigned*)(total + 1);     // 1 uint

  sq_init_kernel<<<(B + 255) / 256, 256, 0, stream>>>(E, sq, hardestU, total, count, B);

  dim3 grid(B / 128, CCHUNKS);
  hardest_neg_kernel<<<grid, 256, 0, stream>>>(E, labels, sq, hardestU, B);

  pos_loss_kernel<<<B, 256, 0, stream>>>(E, labels, sq, hardestU, total, count, B);
  finalize_kernel<<<1, 1, 0, stream>>>(total, count, (float*)d_out);
}